// BranchValueHead_83064667504975
// MI455X (gfx1250) — compile-verified
//
#include <hip/hip_runtime.h>
#include <hip/hip_bf16.h>

typedef __attribute__((ext_vector_type(2))) float v2f;
typedef __attribute__((ext_vector_type(8))) float v8f;

#define CCH   128           // channels
#define KBR   32            // branches per graph
#define NGW   3             // graph window in LDS
#define CHUNK 1536          // nodes per block in stage 1

// ---------------- Stage 0: init accumulator / max_b / out ----------------
__global__ void bvh_init_kernel(float* __restrict__ accum, int* __restrict__ maxb,
                                float* __restrict__ out, int accumElems, int nGraphs) {
    int i = blockIdx.x * 256 + threadIdx.x;
    if (i < accumElems) accum[i] = 0.0f;
    if (i < nGraphs) { maxb[i] = -1; out[i] = 0.0f; }
}

// ---------------- Stage 1: LDS-privatized segment sum + segment max ----------------
__global__ void bvh_scatter_kernel(const float* __restrict__ ne,
                                   const int* __restrict__ batch,
                                   const int* __restrict__ branch,
                                   float* __restrict__ accum,
                                   int* __restrict__ maxb,
                                   int nNodes, int nGraphs) {
    __shared__ float acc[NGW * KBR * CCH];   // 48 KB
    __shared__ int   lmax[NGW];
    __shared__ int   sg0;
    const int tid = threadIdx.x;

    for (int i = tid; i < NGW * KBR * CCH; i += 256) acc[i] = 0.0f;
    if (tid < NGW) lmax[tid] = -1;

    const int s = blockIdx.x * CHUNK;
    const int e = min(s + CHUNK, nNodes);
    if (tid == 0) sg0 = batch[s];
    __syncthreads();
    const int g0 = sg0;

    const int cnt = (e - s) * 32;            // one float4 (4 channels) per work item
    for (int i = tid; i < cnt; i += 256) {
        const int node = s + (i >> 5);
        const int c4   = (i & 31) << 2;
        const int g    = batch[node];
        const int b    = branch[node];
        const float4 v = *(const float4*)(ne + (long long)node * CCH + c4);
        const int w    = g - g0;
        if (w >= 0 && w < NGW) {
            float* p = acc + (w * KBR + b) * CCH + c4;
            atomicAdd(p + 0, v.x); atomicAdd(p + 1, v.y);
            atomicAdd(p + 2, v.z); atomicAdd(p + 3, v.w);
            if ((i & 31) == 0) atomicMax(&lmax[w], b);
        } else {                              // rare fallback (window overflow)
            float* p = accum + ((long long)g * KBR + b) * CCH + c4;
            atomicAdd(p + 0, v.x); atomicAdd(p + 1, v.y);
            atomicAdd(p + 2, v.z); atomicAdd(p + 3, v.w);
            if ((i & 31) == 0) atomicMax(&maxb[g], b);
        }
    }
    __syncthreads();

    // flush LDS window to the global accumulator (skip zeros)
    const long long base = (long long)g0 * KBR * CCH;
    const long long lim  = (long long)nGraphs * KBR * CCH;
    for (int i = tid; i < NGW * KBR * CCH; i += 256) {
        const float v = acc[i];
        const long long gidx = base + i;
        if (v != 0.0f && gidx < lim) atomicAdd(&accum[gidx], v);
    }
    if (tid < NGW) {
        const int mv = lmax[tid];
        const int g  = g0 + tid;
        if (mv >= 0 && g < nGraphs) atomicMax(&maxb[g], mv);
    }
}

// ---------------- Stage 2: WMMA f32 MLP + masked per-graph reduce ----------------
// One wave computes one 16-row tile of branch_embed @ W1 (+b1, leakyReLU, @W2, +b2),
// then reduces its 16 branch values into out[graph] with the b<=max_b mask.
__global__ void bvh_mlp_kernel(const float* __restrict__ E,     // [B*K, 128] accumulator
                               const float* __restrict__ W1,    // [128,128]
                               const float* __restrict__ b1,    // [128]
                               const float* __restrict__ W2,    // [128]
                               const float* __restrict__ b2p,   // [1]
                               const int*  __restrict__ maxb,   // [B]
                               float* __restrict__ out,         // [B]
                               int nTiles) {
    __shared__ float w1s[CCH * CCH];          // 64 KB
    for (int i = threadIdx.x; i < CCH * CCH; i += 256) w1s[i] = W1[i];
    __syncthreads();

    const int wave = threadIdx.x >> 5;
    const int lane = threadIdx.x & 31;
    const int tile = blockIdx.x * 8 + wave;
    if (tile >= nTiles) return;               // wave-uniform: EXEC stays all-1s

    const int m16 = lane & 15;                // A layout: lane -> row M
    const int kb  = (lane >> 4) << 1;         // lanes 0-15 -> K{0,1}, 16-31 -> K{2,3}

    // Preload this wave's 16x128 A slab (2 f32 per lane per K-step of 4)
    const float* Arow = E + (long long)(tile * 16 + m16) * CCH + kb;
    v2f a[32];
#pragma unroll
    for (int ks = 0; ks < 32; ++ks) a[ks] = *(const v2f*)(Arow + ks * 4);

    float s[8];
#pragma unroll
    for (int j = 0; j < 8; ++j) s[j] = 0.0f;

    for (int t = 0; t < 8; ++t) {             // 8 column tiles of 16
        v8f c = {};
#pragma unroll
        for (int ks = 0; ks < 32; ++ks) {     // K = 128 in steps of 4
            v2f bv = { w1s[(ks * 4 + kb) * CCH + t * 16 + m16],
                       w1s[(ks * 4 + kb + 1) * CCH + t * 16 + m16] };
            c = __builtin_amdgcn_wmma_f32_16x16x4_f32(
                    false, a[ks], false, bv, (short)0, c, false, false);
        }
        const int   col  = t * 16 + m16;      // C layout: lane -> column N
        const float bias = b1[col];
        const float w2v  = W2[col];
#pragma unroll
        for (int j = 0; j < 8; ++j) {         // rows j (lanes<16) / j+8 (lanes>=16)
            float h = c[j] + bias;
            h = (h >= 0.0f) ? h : 0.01f * h;  // leaky_relu
            s[j] += h * w2v;
        }
    }

    // XOR butterfly inside each 16-lane half: sum over the 16 columns it holds
#pragma unroll
    for (int j = 0; j < 8; ++j) {
#pragma unroll
        for (int msk = 1; msk < 16; msk <<= 1) s[j] += __shfl_xor(s[j], msk, 32);
    }

    // lanes 0-7 own rows 0-7, lanes 16-23 own rows 8-15
    float v = 0.0f; int m = -1;
#pragma unroll
    for (int j = 0; j < 8; ++j)
        if ((lane & 15) == j) { v = s[j]; m = (lane < 16) ? j : j + 8; }

    if (m >= 0 && lane < 24) {
        const int gid = tile * 16 + m;
        const int g   = gid >> 5;             // graph
        const int b   = gid & (KBR - 1);      // branch slot
        if (b <= maxb[g]) atomicAdd(&out[g], v + b2p[0]);
    }
}

extern "C" void kernel_launch(void* const* d_in, const int* in_sizes, int n_in,
                              void* d_out, int out_size, void* d_ws, size_t ws_size,
                              hipStream_t stream) {
    const float* node_embed = (const float*)d_in[0];
    const int*   batch      = (const int*)d_in[1];
    const int*   branch     = (const int*)d_in[2];
    const float* W1         = (const float*)d_in[3];
    const float* b1         = (const float*)d_in[4];
    const float* W2         = (const float*)d_in[5];
    const float* b2         = (const float*)d_in[6];
    float*       out        = (float*)d_out;

    const int N       = in_sizes[1];           // num nodes
    const int B       = out_size;              // num graphs (out is [B,1])
    const int accumEl = B * KBR * CCH;         // 8192 * 128

    float* accum = (float*)d_ws;
    int*   maxb  = (int*)((char*)d_ws + (size_t)accumEl * sizeof(float));

    // Stage 0: clear accumulator / max_b / out
    {
        int total  = accumEl;
        int blocks = (total + 255) / 256;
        bvh_init_kernel<<<blocks, 256, 0, stream>>>(accum, maxb, out, accumEl, B);
    }
    // Stage 1: segment sum + segment max (memory-bound, 1.02 GB streamed)
    {
        int blocks = (N + CHUNK - 1) / CHUNK;
        bvh_scatter_kernel<<<blocks, 256, 0, stream>>>(node_embed, batch, branch,
                                                       accum, maxb, N, B);
    }
    // Stage 2: WMMA MLP + masked graph reduction
    {
        int nTiles = (B * KBR) / 16;           // 512
        int blocks = (nTiles + 7) / 8;         // 8 waves per block
        bvh_mlp_kernel<<<blocks, 256, 0, stream>>>(accum, W1, b1, W2, b2,
                                                   maxb, out, nTiles);
    }
}